// STEPGraphEncoder_22359599743038
// MI455X (gfx1250) — compile-verified
//
#include <hip/hip_runtime.h>
#include <hip/hip_bf16.h>

typedef __attribute__((ext_vector_type(2))) float v2f;
typedef __attribute__((ext_vector_type(8))) float v8f;

#define NODE_DIM 128

// ---------------------------------------------------------------------------
// WMMA GEMM: out[N,128] = epilogue( A[N,K] @ B[K,128] + bias )
// epilogue: v = relu(acc + bias); out = RESIDUAL ? out + v : v
//
// Block: 256 threads = 8 waves; wave w owns N-tile w (16 cols).
// Each block processes MT M-tiles (MT*16 rows). B fragments for the wave's
// 128x16 weight tile are loaded ONCE into registers (K/4 v2f = 64 VGPRs for
// K=128) and reused across all MT tiles. Inner loop: 1 ds_load_2addr_b64
// feeds 2 v_wmma_f32_16x16x4_f32. Full tiles take unguarded vectorized
// fill/store paths; only the last partial block pays per-row guards.
// ---------------------------------------------------------------------------
template <int K, int MT, bool RESIDUAL>
__global__ void gcn_gemm_wmma_kernel(const float* __restrict__ A,
                                     const float* __restrict__ B,
                                     const float* __restrict__ bias,
                                     float* __restrict__ out,
                                     int Nrows) {
    __shared__ float As[16 * K];  // K <= 128 -> max 8 KB

    const int wave = threadIdx.x >> 5;    // 0..7 -> N tile
    const int lane = threadIdx.x & 31;

    // Fragment lane mapping (f32 16x16x4):
    //  A 16x4: lanes 0-15 -> M=lane, K={k0,k0+1}; lanes 16-31 -> M=lane-16, K={k0+2,k0+3}
    //  B 4x16: lanes 0-15 -> K rows {k0,k0+1}; lanes 16-31 -> {k0+2,k0+3}; N=lane&15
    const int m  = lane & 15;
    const int kl = (lane >> 4) * 2;       // 0 or 2
    const int n  = wave * 16 + (lane & 15);

    // ---- load the wave's B fragments once (register resident) ----
    v2f bf[K / 4];
#pragma unroll
    for (int s = 0; s < K / 4; ++s) {
        const int k0 = s * 4 + kl;
        bf[s].x = B[(long)k0 * NODE_DIM + n];
        bf[s].y = B[(long)(k0 + 1) * NODE_DIM + n];
    }
    const float bs = bias[n];
    const long aMax = (long)Nrows * K;

    for (int mt = 0; mt < MT; ++mt) {
        const long mtile = (long)blockIdx.x * MT + mt;
        const long rowBase = mtile * 16;

        __syncthreads();  // previous tile's LDS reads complete before refill
        const long aBase = rowBase * K;
        if (aBase + 16 * K <= aMax) {
            // full tile: vectorized b128 copy
            const float4* src = (const float4*)(A + aBase);
            float4* dst = (float4*)As;
            for (int idx = threadIdx.x; idx < 16 * K / 4; idx += 256)
                dst[idx] = src[idx];
        } else {
            for (int idx = threadIdx.x; idx < 16 * K; idx += 256) {
                const long g = aBase + idx;
                As[idx] = (g < aMax) ? A[g] : 0.0f;
            }
        }
        __syncthreads();

        v8f acc = {};
#pragma unroll
        for (int s = 0; s < K / 4; ++s) {
            v2f a;
            a.x = As[m * K + s * 4 + kl];
            a.y = As[m * K + s * 4 + kl + 1];
            // 8 args: (neg_a, A, neg_b, B, c_mod, C, reuse_a, reuse_b)
            acc = __builtin_amdgcn_wmma_f32_16x16x4_f32(
                false, a, false, bf[s], (short)0, acc, false, false);
        }

        // D/C layout: VGPR r -> M = r (lanes 0-15) or r+8 (lanes 16-31), N = lane&15
        const int rsel = (lane < 16) ? 0 : 8;
        if (rowBase + 16 <= Nrows) {
            // full tile: unguarded stores
            float* op = out + (rowBase + rsel) * NODE_DIM + n;
#pragma unroll
            for (int r = 0; r < 8; ++r) {
                float v = acc[r] + bs;
                v = v > 0.0f ? v : 0.0f;
                if (RESIDUAL) v += op[(long)r * NODE_DIM];
                op[(long)r * NODE_DIM] = v;
            }
        } else {
#pragma unroll
            for (int r = 0; r < 8; ++r) {
                const long row = rowBase + rsel + r;
                if (row < Nrows) {
                    const long o = row * NODE_DIM + n;
                    float v = acc[r] + bs;
                    v = v > 0.0f ? v : 0.0f;
                    if (RESIDUAL) v += out[o];
                    out[o] = v;
                }
            }
        }
    }
}

// ---------------------------------------------------------------------------
// deg[i] = 1.0 (self-loop contribution)
// ---------------------------------------------------------------------------
__global__ void deg_init_kernel(float* __restrict__ deg, int N) {
    const int i = blockIdx.x * blockDim.x + threadIdx.x;
    if (i < N) deg[i] = 1.0f;
}

// ---------------------------------------------------------------------------
// deg[row[e]] += vals[e]
// ---------------------------------------------------------------------------
__global__ void deg_accum_kernel(const int* __restrict__ row,
                                 const float* __restrict__ vals,
                                 float* __restrict__ deg, int E) {
    const int e = blockIdx.x * blockDim.x + threadIdx.x;
    if (e < E) atomicAdd(&deg[row[e]], vals[e]);
}

// ---------------------------------------------------------------------------
// deg[i] = (max(deg[i],1))^{-1/2}   (in place -> becomes dinv)
// ---------------------------------------------------------------------------
__global__ void dinv_kernel(float* __restrict__ deg, int N) {
    const int i = blockIdx.x * blockDim.x + threadIdx.x;
    if (i < N) {
        float d = deg[i];
        d = d > 1.0f ? d : 1.0f;
        deg[i] = 1.0f / sqrtf(d);
    }
}

// ---------------------------------------------------------------------------
// messages[i][f] = dinv[i]^2 * x[i][f]   (self-loop term, also zero-init)
// ---------------------------------------------------------------------------
__global__ void msg_init_kernel(const float* __restrict__ x,
                                const float* __restrict__ dinv,
                                float* __restrict__ msg, int N) {
    const long t = (long)blockIdx.x * blockDim.x + threadIdx.x;
    if (t < (long)N * NODE_DIM) {
        const int i = (int)(t >> 7);  // t / 128
        const float d = dinv[i];
        msg[t] = d * d * x[t];
    }
}

// ---------------------------------------------------------------------------
// SpMM scatter: one wave (32 lanes) per edge, float4 per lane.
// messages[row[e]][:] += vals[e]*dinv[row]*dinv[col] * x[col[e]][:]
// x and messages are L2-resident (51.2 MB each << 192 MB L2).
// ---------------------------------------------------------------------------
__global__ void spmm_kernel(const int* __restrict__ row,
                            const int* __restrict__ col,
                            const float* __restrict__ vals,
                            const float* __restrict__ dinv,
                            const float* __restrict__ x,
                            float* __restrict__ msg, int E) {
    const long e = (long)blockIdx.x * (blockDim.x >> 5) + (threadIdx.x >> 5);
    if (e >= E) return;
    const int lane = threadIdx.x & 31;
    const int r = row[e];
    const int c = col[e];
    const float w = vals[e] * dinv[r] * dinv[c];
    const float4 v = ((const float4*)(x + (long)c * NODE_DIM))[lane];
    float* m = msg + (long)r * NODE_DIM + lane * 4;
    atomicAdd(m + 0, w * v.x);
    atomicAdd(m + 1, w * v.y);
    atomicAdd(m + 2, w * v.z);
    atomicAdd(m + 3, w * v.w);
}

// ---------------------------------------------------------------------------
extern "C" void kernel_launch(void* const* d_in, const int* in_sizes, int n_in,
                              void* d_out, int out_size, void* d_ws, size_t ws_size,
                              hipStream_t stream) {
    const float* node_features = (const float*)d_in[0];  // [N,48]
    const int*   edge_row      = (const int*)d_in[1];    // [E]
    const int*   edge_col      = (const int*)d_in[2];    // [E]
    const float* edge_vals     = (const float*)d_in[3];  // [E]
    const float* W_in          = (const float*)d_in[4];  // [48,128]
    const float* b_in          = (const float*)d_in[5];  // [128]
    const float* W_conv        = (const float*)d_in[6];  // [3,128,128]
    const float* b_conv        = (const float*)d_in[7];  // [3,128]

    const int IN_DIM = 48;
    const int N = in_sizes[0] / IN_DIM;
    const int E = in_sizes[1];
    const int NUM_LAYERS = in_sizes[7] / NODE_DIM;

    float* x   = (float*)d_out;                  // [N,128], lives in d_out
    float* msg = (float*)d_ws;                   // [N,128]
    float* deg = msg + (long)N * NODE_DIM;       // [N] -> becomes dinv

    constexpr int MT = 8;                        // M-tiles per block (128 rows)
    const int gemm_blocks = (N + MT * 16 - 1) / (MT * 16);

    // 1) input projection: x = relu(nf @ W_in + b_in)
    gcn_gemm_wmma_kernel<48, MT, false><<<gemm_blocks, 256, 0, stream>>>(
        node_features, W_in, b_in, x, N);

    // 2) degree + symmetric normalization
    deg_init_kernel<<<(N + 255) / 256, 256, 0, stream>>>(deg, N);
    deg_accum_kernel<<<(E + 255) / 256, 256, 0, stream>>>(edge_row, edge_vals, deg, E);
    dinv_kernel<<<(N + 255) / 256, 256, 0, stream>>>(deg, N);

    // 3) conv layers
    const long nm = (long)N * NODE_DIM;
    const int msg_blocks  = (int)((nm + 255) / 256);
    const int spmm_blocks = (E + 7) / 8;  // 8 edges (waves) per 256-thread block
    for (int l = 0; l < NUM_LAYERS; ++l) {
        msg_init_kernel<<<msg_blocks, 256, 0, stream>>>(x, deg, msg, N);
        spmm_kernel<<<spmm_blocks, 256, 0, stream>>>(
            edge_row, edge_col, edge_vals, deg, x, msg, E);
        gcn_gemm_wmma_kernel<128, MT, true><<<gemm_blocks, 256, 0, stream>>>(
            msg, W_conv + (long)l * NODE_DIM * NODE_DIM,
            b_conv + (long)l * NODE_DIM, x, N);
    }
}